// GeometricGraphNeuralNetwork_90056874262562
// MI455X (gfx1250) — compile-verified
//
#include <hip/hip_runtime.h>
#include <math.h>

#define Nn 50000
#define Ee 800000

typedef __attribute__((ext_vector_type(2))) float v2f;
typedef __attribute__((ext_vector_type(8))) float v8f;

__device__ __forceinline__ float gelu_f(float x) {
  return 0.5f * x * (1.0f + erff(x * 0.70710678118654752440f));
}
// monotone float<->uint encoding so atomicMax(uint) == float max, memset(0) == -inf
__device__ __forceinline__ unsigned fenc(float f) {
  unsigned u = __float_as_uint(f);
  return (u & 0x80000000u) ? ~u : (u | 0x80000000u);
}
__device__ __forceinline__ float fdec(unsigned e) {
  return __uint_as_float((e & 0x80000000u) ? (e & 0x7fffffffu) : ~e);
}

// ---------------- WMMA f32 linear: Y[n,o] = actO( actI(X[n,:]) @ W[o,:] + b[o] )
// one wave per 16x16 tile, K in steps of 4 via V_WMMA_F32_16X16X4_F32
template <int AIN, int AOUT>
__global__ __launch_bounds__(32) void wmma_linear(
    const float* __restrict__ X, int ldX,
    const float* __restrict__ W, int ldW, int DI,
    const float* __restrict__ bias,
    float* __restrict__ Y, int ldY) {
  const int o0 = blockIdx.x * 16;
  const int n0 = blockIdx.y * 16;
  const int lane = threadIdx.x;
  const int half = lane >> 4;   // 0/1
  const int l16 = lane & 15;
  const int kb = half * 2;      // K sub-offset within group of 4

  const float* xrow = X + (long)(n0 + l16) * ldX;  // A: row n0+l16
  const float* wrow = W + (long)(o0 + l16) * ldW;  // B: col o0+l16 of W^T

  v8f acc = {};
  const int DI4 = DI & ~3;
  for (int k = 0; k < DI4; k += 4) {
    float ax = xrow[k + kb], ay = xrow[k + kb + 1];
    if (AIN == 1) { ax = gelu_f(ax); ay = gelu_f(ay); }
    v2f a, b;
    a.x = ax; a.y = ay;
    b.x = wrow[k + kb]; b.y = wrow[k + kb + 1];
    acc = __builtin_amdgcn_wmma_f32_16x16x4_f32(false, a, false, b, (short)0,
                                                acc, false, false);
  }
  // scalar tail for DI not multiple of 4 (topo layer DI=129)
  for (int k = DI4; k < DI; ++k) {
    float wv = wrow[k];
#pragma unroll
    for (int r = 0; r < 8; ++r) {
      float xv = X[(long)(n0 + r + half * 8) * ldX + k];
      if (AIN == 1) xv = gelu_f(xv);
      acc[r] += xv * wv;
    }
  }
  const float bv = bias ? bias[o0 + l16] : 0.0f;
#pragma unroll
  for (int r = 0; r < 8; ++r) {
    float y = acc[r] + bv;
    if (AOUT == 1) y = gelu_f(y);
    Y[(long)(n0 + r + half * 8) * ldY + o0 + l16] = y;
  }
}

// ---------------- edge / elementwise kernels ----------------
__global__ void k_edge_cagcn(const int* __restrict__ ei, const float* __restrict__ x,
                             const float* __restrict__ curv, const float* __restrict__ cw,
                             const float* __restrict__ cb, float* __restrict__ sum,
                             float* __restrict__ cnt) {
  long i = (long)blockIdx.x * blockDim.x + threadIdx.x;
  if (i >= (long)Ee * 64) return;
  int e = (int)(i >> 6), d = (int)(i & 63);
  int row = ei[e], col = ei[Ee + e];
  float cd = fabsf(curv[row] - curv[col]);
  float g = 1.0f / (1.0f + expf(-(cd * cw[d] + cb[d])));
  atomicAdd(&sum[(long)row * 64 + d], x[(long)col * 64 + d] * g);
  if (d == 0) atomicAdd(&cnt[row], 1.0f);
}

__global__ void k_divcnt(float* __restrict__ s, const float* __restrict__ cnt) {
  long i = (long)blockIdx.x * blockDim.x + threadIdx.x;
  if (i >= (long)Nn * 64) return;
  s[i] /= fmaxf(cnt[i >> 6], 1.0f);
}

__global__ void k_asad(const float* __restrict__ h, const float* __restrict__ asrc,
                       const float* __restrict__ adst, float* __restrict__ as_,
                       float* __restrict__ ad_) {
  int i = blockIdx.x * blockDim.x + threadIdx.x;
  if (i >= Nn * 8) return;
  int n = i >> 3, hh = i & 7;
  float s = 0.f, d = 0.f;
  for (int j = 0; j < 8; ++j) {
    float v = h[(long)n * 64 + hh * 8 + j];
    s += v * asrc[hh * 8 + j];
    d += v * adst[hh * 8 + j];
  }
  as_[i] = s; ad_[i] = d;
}

__global__ void k_edge_logit(const int* __restrict__ ei, const float* __restrict__ ea,
                             const float* __restrict__ gew, const float* __restrict__ as_,
                             const float* __restrict__ ad_, float* __restrict__ eb,
                             unsigned* __restrict__ mx) {
  long i = (long)blockIdx.x * blockDim.x + threadIdx.x;
  if (i >= (long)Ee * 8) return;
  int e = (int)(i >> 3), hh = (int)(i & 7);
  int row = ei[e], col = ei[Ee + e];
  float l = as_[col * 8 + hh] + ad_[row * 8 + hh];
  for (int j = 0; j < 4; ++j) l += ea[(long)e * 4 + j] * gew[j * 8 + hh];
  l = (l > 0.f) ? l : 0.2f * l;  // leaky_relu 0.2
  eb[i] = l;
  atomicMax(&mx[row * 8 + hh], fenc(l));
}

__global__ void k_edge_exp(const int* __restrict__ ei, float* __restrict__ eb,
                           const unsigned* __restrict__ mx, float* __restrict__ den) {
  long i = (long)blockIdx.x * blockDim.x + threadIdx.x;
  if (i >= (long)Ee * 8) return;
  int e = (int)(i >> 3), hh = (int)(i & 7);
  int row = ei[e];
  float ex = expf(eb[i] - fdec(mx[row * 8 + hh]));
  eb[i] = ex;
  atomicAdd(&den[row * 8 + hh], ex);
}

__global__ void k_edge_gat(const int* __restrict__ ei, const float* __restrict__ eb,
                           const float* __restrict__ den, const float* __restrict__ h,
                           float* __restrict__ f2) {
  long i = (long)blockIdx.x * blockDim.x + threadIdx.x;
  if (i >= (long)Ee * 64) return;
  int e = (int)(i >> 6), d = (int)(i & 63);
  int row = ei[e], col = ei[Ee + e], hh = d >> 3;
  float att = eb[(long)e * 8 + hh] / fmaxf(den[row * 8 + hh], 1e-16f);
  atomicAdd(&f2[(long)row * 64 + d], h[(long)col * 64 + d] * att);
}

__global__ void k_elu(float* __restrict__ f) {
  long i = (long)blockIdx.x * blockDim.x + threadIdx.x;
  if (i >= (long)Nn * 64) return;
  float v = f[i];
  f[i] = (v > 0.f) ? v : expm1f(v);
}

__global__ void k_edge_spec(const int* __restrict__ ei, const float* __restrict__ f2,
                            const float* __restrict__ cnt, float* __restrict__ s3) {
  long i = (long)blockIdx.x * blockDim.x + threadIdx.x;
  if (i >= (long)Ee * 64) return;
  int e = (int)(i >> 6), d = (int)(i & 63);
  int row = ei[e], col = ei[Ee + e];
  float w = rsqrtf(fmaxf(cnt[col], 1.f)) * rsqrtf(fmaxf(cnt[row], 1.f));
  atomicAdd(&s3[(long)row * 64 + d], f2[(long)col * 64 + d] * w);
}

__global__ void k_xo(const float* __restrict__ f1, const float* __restrict__ f2,
                     const float* __restrict__ f3, float* __restrict__ c) {
  long i = (long)blockIdx.x * blockDim.x + threadIdx.x;
  if (i >= (long)Nn * 64) return;
  int n = (int)(i >> 6), d = (int)(i & 63);
  c[(long)n * 96 + d] = (f1[i] + f2[i] + f3[i]) * (1.f / 3.f);
}

__global__ void k_edge_topo(const int* __restrict__ ei, const float* __restrict__ c,
                            float* __restrict__ xs) {
  long i = (long)blockIdx.x * blockDim.x + threadIdx.x;
  if (i >= (long)Ee * 64) return;
  int e = (int)(i >> 6), d = (int)(i & 63);
  int row = ei[e], col = ei[Ee + e];
  atomicAdd(&xs[(long)row * 64 + d], c[(long)col * 96 + d]);
}

__global__ void k_build_t(const float* __restrict__ c, const float* __restrict__ xs,
                          const float* __restrict__ cnt, float* __restrict__ t) {
  long i = (long)blockIdx.x * blockDim.x + threadIdx.x;
  if (i >= (long)Nn * 129) return;
  int n = (int)(i / 129), j = (int)(i % 129);
  float v;
  if (j < 64) v = c[(long)n * 96 + j];
  else if (j < 128) v = xs[(long)n * 64 + (j - 64)] / fmaxf(cnt[n], 1.f);
  else v = log1pf(cnt[n]);
  t[i] = v;
}

__global__ void k_pack(const float* __restrict__ c1, const float* __restrict__ c2,
                       float* __restrict__ s) {
  long i = (long)blockIdx.x * blockDim.x + threadIdx.x;
  if (i >= (long)2 * Nn * 96) return;
  int r = (int)(i / 96), f = (int)(i % 96), n = r >> 1;
  s[i] = (r & 1) ? c2[(long)n * 96 + f] : c1[(long)n * 96 + f];
}

__global__ void k_attn(const float* __restrict__ qkv, float* __restrict__ o2) {
  int i = blockIdx.x * blockDim.x + threadIdx.x;
  if (i >= Nn * 4) return;
  int n = i >> 2, m = i & 3;
  const float* r0 = qkv + (long)(2 * n) * 288;
  const float* r1 = r0 + 288;
  int qo = m * 24, ko = 96 + m * 24, vo = 192 + m * 24;
  float s00 = 0, s01 = 0, s10 = 0, s11 = 0;
  for (int d = 0; d < 24; ++d) {
    float q0 = r0[qo + d], q1 = r1[qo + d];
    float k0 = r0[ko + d], k1 = r1[ko + d];
    s00 += q0 * k0; s01 += q0 * k1; s10 += q1 * k0; s11 += q1 * k1;
  }
  const float sc = 0.20412414523193150818f;  // 1/sqrt(24)
  s00 *= sc; s01 *= sc; s10 *= sc; s11 *= sc;
  float m0 = fmaxf(s00, s01), m1 = fmaxf(s10, s11);
  float e00 = expf(s00 - m0), e01 = expf(s01 - m0);
  float e10 = expf(s10 - m1), e11 = expf(s11 - m1);
  float a00 = e00 / (e00 + e01), a01 = e01 / (e00 + e01);
  float a10 = e10 / (e10 + e11), a11 = e11 / (e10 + e11);
  float* w0 = o2 + (long)(2 * n) * 96 + m * 24;
  float* w1 = w0 + 96;
  for (int d = 0; d < 24; ++d) {
    float v0 = r0[vo + d], v1 = r1[vo + d];
    w0[d] = a00 * v0 + a01 * v1;
    w1[d] = a10 * v0 + a11 * v1;
  }
}

__global__ void k_mean(const float* __restrict__ p, float* __restrict__ fo) {
  long i = (long)blockIdx.x * blockDim.x + threadIdx.x;
  if (i >= (long)Nn * 96) return;
  int n = (int)(i / 96), f = (int)(i % 96);
  fo[i] = 0.5f * (p[(long)(2 * n) * 96 + f] + p[(long)(2 * n + 1) * 96 + f]);
}

__global__ void k_ln(float* __restrict__ h, const float* __restrict__ g,
                     const float* __restrict__ b) {
  int n = blockIdx.x * blockDim.x + threadIdx.x;
  if (n >= Nn) return;
  float* r = h + (long)n * 64;
  float mu = 0.f;
  for (int d = 0; d < 64; ++d) mu += r[d];
  mu *= (1.f / 64.f);
  float var = 0.f;
  for (int d = 0; d < 64; ++d) { float t = r[d] - mu; var += t * t; }
  var *= (1.f / 64.f);
  float inv = rsqrtf(var + 1e-5f);
  for (int d = 0; d < 64; ++d) r[d] = (r[d] - mu) * inv * g[d] + b[d];
}

__global__ void k_proj(const float* __restrict__ h2, const float* __restrict__ w3,
                       const float* __restrict__ b3, float* __restrict__ out) {
  int n = blockIdx.x * blockDim.x + threadIdx.x;
  if (n >= Nn) return;
  float s = b3[0];
  for (int j = 0; j < 32; ++j) s += h2[(long)n * 32 + j] * w3[j];
  out[n] = s;
}

// ---------------- host orchestration ----------------
static inline int nb(long total) { return (int)((total + 255) / 256); }

static void run_embed(const float* x, const int* ei, const float* ea, const float* curv,
                      const float* cagcn_w, const float* cagcn_b, const float* curv_w,
                      const float* curv_b, const float* gat_w, const float* gat_b,
                      const float* a_src, const float* a_dst, const float* gew,
                      const float* spec_w, const float* spec_b, const float* tw1,
                      const float* tb1, const float* tw2, const float* tb2,
                      float* cout, float* pool, hipStream_t s) {
  const long N64 = (long)Nn * 64, N8 = (long)Nn * 8, E8 = (long)Ee * 8;
  float* n64a = pool;
  float* n64b = n64a + N64;
  float* n64c = n64b + N64;
  float* n64d = n64c + N64;
  float* tb = n64d + N64;          // N*129
  float* eb = tb + (long)Nn * 129; // E*8
  float* asb = eb + E8;
  float* adb = asb + N8;
  float* den = adb + N8;
  unsigned* mx = (unsigned*)(den + N8);
  float* cnt = (float*)(mx + N8);

  hipMemsetAsync(n64a, 0, N64 * 4, s);
  hipMemsetAsync(cnt, 0, Nn * 4, s);
  k_edge_cagcn<<<nb((long)Ee * 64), 256, 0, s>>>(ei, x, curv, curv_w, curv_b, n64a, cnt);
  k_divcnt<<<nb(N64), 256, 0, s>>>(n64a, cnt);
  wmma_linear<0, 1><<<dim3(4, Nn / 16), 32, 0, s>>>(n64a, 64, cagcn_w, 64, 64, cagcn_b, n64b, 64);  // f1
  wmma_linear<0, 0><<<dim3(4, Nn / 16), 32, 0, s>>>(n64b, 64, gat_w, 64, 64, gat_b, n64c, 64);      // h
  k_asad<<<nb(N8), 256, 0, s>>>(n64c, a_src, a_dst, asb, adb);
  hipMemsetAsync(mx, 0, N8 * 4, s);   // fenc(-inf)-compatible
  hipMemsetAsync(den, 0, N8 * 4, s);
  k_edge_logit<<<nb(E8), 256, 0, s>>>(ei, ea, gew, asb, adb, eb, mx);
  k_edge_exp<<<nb(E8), 256, 0, s>>>(ei, eb, mx, den);
  hipMemsetAsync(n64a, 0, N64 * 4, s);
  k_edge_gat<<<nb((long)Ee * 64), 256, 0, s>>>(ei, eb, den, n64c, n64a);
  k_elu<<<nb(N64), 256, 0, s>>>(n64a);  // f2
  hipMemsetAsync(n64c, 0, N64 * 4, s);
  k_edge_spec<<<nb((long)Ee * 64), 256, 0, s>>>(ei, n64a, cnt, n64c);
  wmma_linear<0, 1><<<dim3(4, Nn / 16), 32, 0, s>>>(n64c, 64, spec_w, 64, 64, spec_b, n64d, 64);    // f3
  k_xo<<<nb(N64), 256, 0, s>>>(n64b, n64a, n64d, cout);
  hipMemsetAsync(n64b, 0, N64 * 4, s);
  k_edge_topo<<<nb((long)Ee * 64), 256, 0, s>>>(ei, cout, n64b);
  k_build_t<<<nb((long)Nn * 129), 256, 0, s>>>(cout, n64b, cnt, tb);
  wmma_linear<0, 1><<<dim3(4, Nn / 16), 32, 0, s>>>(tb, 129, tw1, 129, 129, tb1, n64c, 64);
  wmma_linear<0, 0><<<dim3(2, Nn / 16), 32, 0, s>>>(n64c, 64, tw2, 64, 64, tb2, cout + 64, 96);
}

extern "C" void kernel_launch(void* const* d_in, const int* in_sizes, int n_in,
                              void* d_out, int out_size, void* d_ws, size_t ws_size,
                              hipStream_t stream) {
  const float* x1 = (const float*)d_in[0];
  const int* ei1 = (const int*)d_in[1];
  const float* ea1 = (const float*)d_in[2];
  const float* cv1 = (const float*)d_in[3];
  const float* x2 = (const float*)d_in[4];
  const int* ei2 = (const int*)d_in[5];
  const float* ea2 = (const float*)d_in[6];
  const float* cv2 = (const float*)d_in[7];
  const float* cagcn_w = (const float*)d_in[8];
  const float* cagcn_b = (const float*)d_in[9];
  const float* curv_w = (const float*)d_in[10];
  const float* curv_b = (const float*)d_in[11];
  const float* gat_w = (const float*)d_in[12];
  const float* gat_b = (const float*)d_in[13];
  const float* a_src = (const float*)d_in[14];
  const float* a_dst = (const float*)d_in[15];
  const float* gew = (const float*)d_in[16];
  const float* spec_w = (const float*)d_in[17];
  const float* spec_b = (const float*)d_in[18];
  const float* tw1 = (const float*)d_in[19];
  const float* tb1 = (const float*)d_in[20];
  const float* tw2 = (const float*)d_in[21];
  const float* tb2 = (const float*)d_in[22];
  const float* mha_in_w = (const float*)d_in[23];
  const float* mha_in_b = (const float*)d_in[24];
  const float* mha_out_w = (const float*)d_in[25];
  const float* mha_out_b = (const float*)d_in[26];
  const float* ch_w1 = (const float*)d_in[27];
  const float* ch_b1 = (const float*)d_in[28];
  const float* ln_g = (const float*)d_in[29];
  const float* ln_b = (const float*)d_in[30];
  const float* ch_w2 = (const float*)d_in[31];
  const float* ch_b2 = (const float*)d_in[32];
  const float* ch_w3 = (const float*)d_in[33];
  const float* ch_b3 = (const float*)d_in[34];
  (void)in_sizes; (void)n_in; (void)out_size; (void)ws_size;

  const long N96 = (long)Nn * 96;
  float* w = (float*)d_ws;
  float* c1 = w;
  float* c2 = c1 + N96;
  float* pool = c2 + N96;

  run_embed(x1, ei1, ea1, cv1, cagcn_w, cagcn_b, curv_w, curv_b, gat_w, gat_b,
            a_src, a_dst, gew, spec_w, spec_b, tw1, tb1, tw2, tb2, c1, pool, stream);
  run_embed(x2, ei2, ea2, cv2, cagcn_w, cagcn_b, curv_w, curv_b, gat_w, gat_b,
            a_src, a_dst, gew, spec_w, spec_b, tw1, tb1, tw2, tb2, c2, pool, stream);

  // MHA stage (pool reused; S aliases O2 region — S dead once qkv computed)
  float* qkv = pool;                       // 2N*288
  float* o2 = qkv + (long)2 * Nn * 288;    // 2N*96 (also packed S)
  float* p2 = o2 + (long)2 * Nn * 96;      // 2N*96
  float* hcl = p2 + (long)2 * Nn * 96;     // N*64
  float* h2 = hcl + (long)Nn * 64;         // N*32
  float* fout = (float*)d_out;             // final [N,96]
  float* pout = fout + N96;                // proj  [N,1]

  k_pack<<<nb((long)2 * Nn * 96), 256, 0, stream>>>(c1, c2, o2);
  wmma_linear<0, 0><<<dim3(18, 2 * Nn / 16), 32, 0, stream>>>(o2, 96, mha_in_w, 96, 96, mha_in_b, qkv, 288);
  k_attn<<<nb(Nn * 4), 256, 0, stream>>>(qkv, o2);
  wmma_linear<0, 0><<<dim3(6, 2 * Nn / 16), 32, 0, stream>>>(o2, 96, mha_out_w, 96, 96, mha_out_b, p2, 96);
  k_mean<<<nb(N96), 256, 0, stream>>>(p2, fout);
  wmma_linear<0, 0><<<dim3(4, Nn / 16), 32, 0, stream>>>(fout, 96, ch_w1, 96, 96, ch_b1, hcl, 64);
  k_ln<<<nb(Nn), 256, 0, stream>>>(hcl, ln_g, ln_b);
  wmma_linear<1, 1><<<dim3(2, Nn / 16), 32, 0, stream>>>(hcl, 64, ch_w2, 64, 64, ch_b2, h2, 32);
  k_proj<<<nb(Nn), 256, 0, stream>>>(h2, ch_w3, ch_b3, pout);
}